// Seq2Seq_42949672960722
// MI455X (gfx1250) — compile-verified
//
#include <hip/hip_runtime.h>
#include <math.h>

// ---- problem constants ----
constexpr int kB  = 32;    // batch
constexpr int kT  = 400;   // encoder length
constexpr int kH  = 512;   // hidden / embed dim
constexpr int kL  = 100;   // decode steps (L-1)
constexpr int kNV = 511;   // fc output dim (VOCAB-1)
constexpr int kG  = 1536;  // 3*kH gate width
constexpr int kLdsStride = 514;  // kH+2: row offsets differ by 2 banks -> no conflicts

typedef __attribute__((ext_vector_type(2))) float v2f;
typedef __attribute__((ext_vector_type(8))) float v8f;

__device__ __forceinline__ v8f wmma4(v2f a, v2f b, v8f c) {
  // D = A(16x4,f32) * B(4x16,f32) + C(16x16,f32)
  return __builtin_amdgcn_wmma_f32_16x16x4_f32(false, a, false, b, (short)0, c,
                                               false, false);
}

__device__ __forceinline__ float sigmoidf_(float v) {
  return 1.0f / (1.0f + expf(-v));
}

// ---------------------------------------------------------------------------
// init: zero hx / sx / ax carries
__global__ void init_kernel(float* hx, float* sx, float* ax) {
  int idx = blockIdx.x * 256 + threadIdx.x;           // 16384 threads
  if (idx < kB * kH) { hx[idx] = 0.0f; sx[idx] = 0.0f; }
  if (idx < kB * kT) { ax[idx] = 0.0f; }
}

// ---------------------------------------------------------------------------
// gi = ix @ W_ih^T ; gh = hx @ W_hh^T  where ix = emb[y[:,step]] (+ sx, step>0)
// one wave per 16x16 tile: 2(M) x 96(N) x 2(matrix) = 384 waves, 48 blocks.
// All 8 waves of a block share (which, mt) -> stage the 16x512 A tile in LDS.
__global__ void gates_kernel(const int* __restrict__ y,
                             const float* __restrict__ emb,
                             const float* __restrict__ sx,
                             const float* __restrict__ hx,
                             const float* __restrict__ Wih,
                             const float* __restrict__ Whh,
                             float* __restrict__ gi, float* __restrict__ gh,
                             int step) {
  __shared__ float sA[16 * kLdsStride];
  const int tid = threadIdx.x;
  const int gw0 = blockIdx.x * 8;
  const int which = gw0 / 192;                 // 0: gi, 1: gh (block-uniform)
  const int mt = (gw0 - which * 192) / 96;     // block-uniform (96 % 8 == 0)

  // ---- cooperative A-tile staging (4096 v2f elements) ----
  if (which == 0) {
#pragma unroll
    for (int i = 0; i < 16; ++i) {
      int idx = tid + 256 * i;                 // 0..4095
      int row = idx >> 8;                      // 256 v2f per row
      int c2  = (idx & 255) * 2;
      int tok = y[(mt * 16 + row) * 101 + step];
      v2f v = *(const v2f*)(emb + (size_t)tok * kH + c2);
      if (step > 0) v += *(const v2f*)(sx + (size_t)(mt * 16 + row) * kH + c2);
      *(v2f*)(sA + row * kLdsStride + c2) = v;
    }
  } else {
#pragma unroll
    for (int i = 0; i < 16; ++i) {
      int idx = tid + 256 * i;
      int row = idx >> 8;
      int c2  = (idx & 255) * 2;
      v2f v = *(const v2f*)(hx + (size_t)(mt * 16 + row) * kH + c2);
      *(v2f*)(sA + row * kLdsStride + c2) = v;
    }
  }
  __syncthreads();

  const int gw   = gw0 + (tid >> 5);
  const int lane = tid & 31;
  const int nt   = (gw - which * 192) - mt * 96;   // 0..95
  const int n0   = nt * 16;

  const float* W = which ? Whh : Wih;        // [1536][512] row-major
  float*       G = which ? gh : gi;          // [32][1536]

  const int khalf = (lane >> 4) * 2;         // lanes 0-15: K{0,1}; 16-31: K{2,3}
  const float* aptr = sA + (lane & 15) * kLdsStride + khalf;
  const float* bptr = W + (size_t)(n0 + (lane & 15)) * kH + khalf;

  v8f acc = {};
#pragma unroll 4
  for (int k0 = 0; k0 < kH; k0 += 4) {
    v2f a = *(const v2f*)(aptr + k0);
    v2f b = *(const v2f*)(bptr + k0);
    acc = wmma4(a, b, acc);
  }
#pragma unroll
  for (int i = 0; i < 8; ++i) {
    int crow = mt * 16 + i + ((lane >> 4) << 3);   // VGPR i -> rows i / i+8
    int ccol = n0 + (lane & 15);
    G[(size_t)crow * kG + ccol] = acc[i];
  }
}

// ---------------------------------------------------------------------------
// GRU gate nonlinearity; hx updated in place (each thread owns one element)
__global__ void gru_combine_kernel(const float* __restrict__ gi,
                                   const float* __restrict__ gh,
                                   const float* __restrict__ bih,
                                   const float* __restrict__ bhh,
                                   float* __restrict__ hx) {
  int idx = blockIdx.x * 256 + threadIdx.x;           // 16384 threads
  int b = idx >> 9, h = idx & (kH - 1);
  const float* gib = gi + (size_t)b * kG;
  const float* ghb = gh + (size_t)b * kG;
  float i_r = gib[h]          + bih[h];
  float i_z = gib[kH + h]     + bih[kH + h];
  float i_n = gib[2*kH + h]   + bih[2*kH + h];
  float h_r = ghb[h]          + bhh[h];
  float h_z = ghb[kH + h]     + bhh[kH + h];
  float h_n = ghb[2*kH + h]   + bhh[2*kH + h];
  float rr = sigmoidf_(i_r + h_r);
  float zz = sigmoidf_(i_z + h_z);
  float nn = tanhf(i_n + rr * h_n);
  float hprev = hx[idx];
  hx[idx] = (1.0f - zz) * nn + zz * hprev;
}

// ---------------------------------------------------------------------------
// score[b,t] = sum_h relu(x[b,t,h] + hx[b,h] + conv?) * aw[h] + ab
// grid: 1600 blocks; block covers 8 consecutive t of one batch row b.
// conv weights (512x15, 30KB) staged in LDS; h*15+k is bank-conflict-free.
__global__ void attn_score_kernel(const float* __restrict__ x,
                                  const float* __restrict__ hx,
                                  const float* __restrict__ ax,
                                  const float* __restrict__ cw,   // [512][15]
                                  const float* __restrict__ cb,
                                  const float* __restrict__ aw,
                                  const float* __restrict__ ab,
                                  float* __restrict__ score, int use_conv) {
  __shared__ float s_cw[kH * 15];
  const int tid = threadIdx.x;
  if (use_conv) {
    for (int i = tid; i < kH * 15; i += 256) s_cw[i] = cw[i];
  }
  __syncthreads();

  const int b = blockIdx.x / 50;
  const int t = (blockIdx.x % 50) * 8 + (tid >> 5);
  const int lane = tid & 31;

  float axv[15];
  if (use_conv) {
#pragma unroll
    for (int k = 0; k < 15; ++k) {
      int tt = t + k - 7;                        // SAME padding, pad=7
      axv[k] = (tt >= 0 && tt < kT) ? ax[b * kT + tt] : 0.0f;
    }
  }

  const float* xrow = x + ((size_t)b * kT + t) * kH;
  const float* hxb  = hx + (size_t)b * kH;
  float part = 0.0f;
#pragma unroll 4
  for (int i = 0; i < 16; ++i) {
    int h = i * 32 + lane;                       // coalesced across lanes
    float v = xrow[h] + hxb[h];
    if (use_conv) {
      float cv = cb[h];
#pragma unroll
      for (int k = 0; k < 15; ++k) cv = fmaf(axv[k], s_cw[h * 15 + k], cv);
      v += cv;
    }
    part += fmaxf(v, 0.0f) * aw[h];
  }
#pragma unroll
  for (int off = 16; off >= 1; off >>= 1) part += __shfl_xor(part, off, 32);
  if (lane == 0) score[b * kT + t] = part + ab[0];
}

// ---------------------------------------------------------------------------
// per-batch softmax over T; writes normalized alignment to ax carry + aligns out
__global__ void softmax_kernel(const float* __restrict__ score,
                               float* __restrict__ ax,
                               float* __restrict__ aligns, int step) {
  __shared__ float sred[16];
  const int b = blockIdx.x, tid = threadIdx.x;
  const int lane = tid & 31, wid = tid >> 5;

  // --- max ---
  float m = -3.402823e38f;
  for (int t = tid; t < kT; t += 256) m = fmaxf(m, score[b * kT + t]);
#pragma unroll
  for (int off = 16; off >= 1; off >>= 1) m = fmaxf(m, __shfl_xor(m, off, 32));
  if (lane == 0) sred[wid] = m;
  __syncthreads();
  if (tid == 0) {
    float mm = sred[0];
    for (int i = 1; i < 8; ++i) mm = fmaxf(mm, sred[i]);
    sred[8] = mm;
  }
  __syncthreads();
  const float gmax = sred[8];

  // --- exp + sum (each thread covers 2 t values; keep them in registers) ---
  float e0 = 0.0f, e1 = 0.0f;
  int t0 = tid, t1 = tid + 256;
  e0 = expf(score[b * kT + t0] - gmax);
  if (t1 < kT) e1 = expf(score[b * kT + t1] - gmax);
  float s = e0 + e1;
#pragma unroll
  for (int off = 16; off >= 1; off >>= 1) s += __shfl_xor(s, off, 32);
  if (lane == 0) sred[wid] = s;
  __syncthreads();
  if (tid == 0) {
    float ss = 0.0f;
    for (int i = 0; i < 8; ++i) ss += sred[i];
    sred[9] = ss;
  }
  __syncthreads();
  const float inv = 1.0f / sred[9];

  float* alg = aligns + ((size_t)b * kL + step) * kT;
  float a0 = e0 * inv;
  ax[b * kT + t0] = a0;
  alg[t0] = a0;
  if (t1 < kT) {
    float a1 = e1 * inv;
    ax[b * kT + t1] = a1;
    alg[t1] = a1;
  }
}

// ---------------------------------------------------------------------------
// partial context: block (b, chunk) accumulates 50 t-rows of a[t] * x[b,t,:]
__global__ void ctx_partial_kernel(const float* __restrict__ x,
                                   const float* __restrict__ ax,
                                   float* __restrict__ partial) {
  const int b = blockIdx.x >> 3;
  const int c = blockIdx.x & 7;
  const int tid = threadIdx.x;
  const float* xb = x + (size_t)b * kT * kH;
  const float* ab = ax + b * kT;
  const int t0 = c * 50, t1 = t0 + 50;
  float acc0 = 0.0f, acc1 = 0.0f;
  for (int t = t0; t < t1; ++t) {
    __builtin_prefetch(xb + (size_t)(t + 2) * kH + tid, 0, 1);
    float av = ab[t];
    acc0 = fmaf(xb[(size_t)t * kH + tid],       av, acc0);
    acc1 = fmaf(xb[(size_t)t * kH + tid + 256], av, acc1);
  }
  float* p = partial + ((size_t)b * 8 + c) * kH;
  p[tid]       = acc0;
  p[tid + 256] = acc1;
}

// sx[b,h] = sum_c partial[b,c,h]
__global__ void ctx_reduce_kernel(const float* __restrict__ partial,
                                  float* __restrict__ sx) {
  int idx = blockIdx.x * 256 + threadIdx.x;     // 16384 threads
  int b = idx >> 9, h = idx & (kH - 1);
  const float* p = partial + (size_t)b * 8 * kH + h;
  float s = 0.0f;
#pragma unroll
  for (int c = 0; c < 8; ++c) s += p[c * kH];
  sx[idx] = s;
}

// ---------------------------------------------------------------------------
// out[:, step, :] = (hx + sx) @ fc_w^T + fc_b   (32 x 511), WMMA f32
// 2(M) x 32(N) = 64 wave-tiles, 8 blocks; A=(hx+sx) tile staged in LDS.
__global__ void fc_kernel(const float* __restrict__ hx,
                          const float* __restrict__ sx,
                          const float* __restrict__ fw,   // [511][512]
                          const float* __restrict__ fb,
                          float* __restrict__ out, int step) {
  __shared__ float sA[16 * kLdsStride];
  const int tid = threadIdx.x;
  const int gw0 = blockIdx.x * 8;
  const int mt  = gw0 >> 5;                    // block-uniform (4 blocks per mt)

  // stage A = hx + sx for rows mt*16 .. mt*16+15
#pragma unroll
  for (int i = 0; i < 16; ++i) {
    int idx = tid + 256 * i;                   // 0..4095 v2f units
    int row = idx >> 8;
    int c2  = (idx & 255) * 2;
    size_t off = (size_t)(mt * 16 + row) * kH + c2;
    v2f v = *(const v2f*)(hx + off);
    v += *(const v2f*)(sx + off);
    *(v2f*)(sA + row * kLdsStride + c2) = v;
  }
  __syncthreads();

  const int gw   = gw0 + (tid >> 5);
  const int lane = tid & 31;
  const int nt   = gw & 31;
  const int n0   = nt * 16;

  const int khalf = (lane >> 4) * 2;
  const int ncol  = n0 + (lane & 15);
  const int nld   = ncol < kNV ? ncol : kNV - 1;   // clamp pad column's load

  const float* aptr = sA + (lane & 15) * kLdsStride + khalf;
  const float* wptr = fw + (size_t)nld * kH + khalf;

  v8f acc = {};
#pragma unroll 4
  for (int k0 = 0; k0 < kH; k0 += 4) {
    v2f a = *(const v2f*)(aptr + k0);
    v2f b = *(const v2f*)(wptr + k0);
    acc = wmma4(a, b, acc);
  }
#pragma unroll
  for (int i = 0; i < 8; ++i) {
    int r = mt * 16 + i + ((lane >> 4) << 3);
    int c = n0 + (lane & 15);
    if (c < kNV)
      out[((size_t)r * kL + step) * kNV + c] = acc[i] + fb[c];
  }
}

// ---------------------------------------------------------------------------
extern "C" void kernel_launch(void* const* d_in, const int* in_sizes, int n_in,
                              void* d_out, int out_size, void* d_ws, size_t ws_size,
                              hipStream_t stream) {
  const float* x   = (const float*)d_in[0];
  const int*   y   = (const int*)  d_in[1];
  const float* emb = (const float*)d_in[2];
  const float* Wih = (const float*)d_in[3];
  const float* Whh = (const float*)d_in[4];
  const float* bih = (const float*)d_in[5];
  const float* bhh = (const float*)d_in[6];
  const float* cw  = (const float*)d_in[7];
  const float* cb  = (const float*)d_in[8];
  const float* aw  = (const float*)d_in[9];
  const float* ab  = (const float*)d_in[10];
  const float* fw  = (const float*)d_in[11];
  const float* fb  = (const float*)d_in[12];

  float* out    = (float*)d_out;                       // [32][100][511]
  float* aligns = out + (size_t)kB * kL * kNV;         // [32][100][400]

  float* ws = (float*)d_ws;
  float* hx = ws;                  // 32*512   = 16384
  float* sx = ws + 16384;          // 32*512
  float* gi = ws + 32768;          // 32*1536  = 49152
  float* gh = ws + 81920;          // 32*1536
  float* ax = ws + 131072;         // 32*400   = 12800
  float* sc = ws + 143872;         // 32*400
  float* pc = ws + 156672;         // 32*8*512 = 131072 (partial contexts)

  init_kernel<<<64, 256, 0, stream>>>(hx, sx, ax);

  for (int step = 0; step < kL; ++step) {
    gates_kernel<<<48, 256, 0, stream>>>(y, emb, sx, hx, Wih, Whh, gi, gh, step);
    gru_combine_kernel<<<64, 256, 0, stream>>>(gi, gh, bih, bhh, hx);
    attn_score_kernel<<<1600, 256, 0, stream>>>(x, hx, ax, cw, cb, aw, ab, sc,
                                                step > 0 ? 1 : 0);
    softmax_kernel<<<32, 256, 0, stream>>>(sc, ax, aligns, step);
    ctx_partial_kernel<<<256, 256, 0, stream>>>(x, ax, pc);
    ctx_reduce_kernel<<<64, 256, 0, stream>>>(pc, sx);
    fc_kernel<<<8, 256, 0, stream>>>(hx, sx, fw, fb, out, step);
  }
}